// MultiBitSTEQuantizer_4724464025934
// MI455X (gfx1250) — compile-verified
//
#include <hip/hip_runtime.h>

// MultiBitSTEQuantizer (BITS=2, CLIP=1.0) forward pass for MI455X (gfx1250).
//
// Memory-bound streaming kernel: 256MB in + 256MB out @ 23.3 TB/s -> ~22us floor.
// Strategy: double-buffered async global->LDS DMA (CDNA5 ASYNCcnt path) feeding
// ds_load_b128 + a tiny branchless 4-level nearest-codebook select, stored with
// non-temporal b128 stores (512MB stream > 192MB L2, so NT avoids thrash).

typedef float v4f __attribute__((ext_vector_type(4)));
typedef int   v4i __attribute__((vector_size(16)));   // matches builtin's V4i

#define TPB   256                    // 8 wave32 per block
#define UNR   4                      // float4 chunks per thread per stage
#define CHUNK (TPB * 4 * UNR)        // 4096 floats per block-stage

#if defined(__has_builtin)
#if __has_builtin(__builtin_amdgcn_global_load_async_to_lds_b128)
#define HAVE_ASYNC_LDS 1
#endif
#endif
#ifndef HAVE_ASYNC_LDS
#define HAVE_ASYNC_LDS 0
#endif

#if defined(__has_builtin) && __has_builtin(__builtin_amdgcn_s_wait_asynccnt)
#define WAIT_ASYNCCNT(n) __builtin_amdgcn_s_wait_asynccnt(n)
#else
#define WAIT_ASYNCCNT(n) asm volatile("s_wait_asynccnt %0" ::"i"(n) : "memory")
#endif

// clamp -> nearest of 4 levels (first-index-wins on ties, matching jnp.argmin)
// -> faithful STE forward arithmetic xc + (q - xc).
// NaN inputs: compares are false -> xc stays NaN, all selects keep q = l0,
// matching argmin-over-all-NaN-distances = index 0.
__device__ __forceinline__ float steq(float x, float l0, float l1, float l2, float l3) {
  float xc = x;
  xc = (xc < -1.0f) ? -1.0f : xc;
  xc = (xc >  1.0f) ?  1.0f : xc;
  float d0 = __builtin_fabsf(xc - l0);
  float d1 = __builtin_fabsf(xc - l1);
  float d2 = __builtin_fabsf(xc - l2);
  float d3 = __builtin_fabsf(xc - l3);
  float q = l0, b = d0;
  q = (d1 < b) ? l1 : q;  b = (d1 < b) ? d1 : b;
  q = (d2 < b) ? l2 : q;  b = (d2 < b) ? d2 : b;
  q = (d3 < b) ? l3 : q;
  return xc + (q - xc);
}

__device__ __forceinline__ v4f steq4(v4f v, float l0, float l1, float l2, float l3) {
  v4f q;
  q.x = steq(v.x, l0, l1, l2, l3);
  q.y = steq(v.y, l0, l1, l2, l3);
  q.z = steq(v.z, l0, l1, l2, l3);
  q.w = steq(v.w, l0, l1, l2, l3);
  return q;
}

#if HAVE_ASYNC_LDS
// Double-buffered async global->LDS pipeline. Each lane DMA-loads its own
// 16B slots and reads only those back, so no workgroup barriers are needed:
// per-wave ASYNCcnt ordering (async loads complete in order) is sufficient.
__global__ __launch_bounds__(TPB) void steq_async_kernel(
    const float* __restrict__ x, const float* __restrict__ levels,
    float* __restrict__ out, long long n_chunks) {
  __shared__ v4f lds[2 * TPB * UNR];  // 2 x 16KB buffers = 32KB
  const int tid = threadIdx.x;
  const float l0 = levels[0], l1 = levels[1], l2 = levels[2], l3 = levels[3];

  long long c = blockIdx.x;
  const long long cstride = gridDim.x;
  int buf = 0;

  auto issue = [&](long long chunk, int b) {
    const float* gbase = x + chunk * (long long)CHUNK;
#pragma unroll
    for (int u = 0; u < UNR; ++u) {
      const float* g = gbase + (u * TPB + tid) * 4;
      v4f* l = &lds[b * (TPB * UNR) + u * TPB + tid];
      __builtin_amdgcn_global_load_async_to_lds_b128(
          (__attribute__((address_space(1))) v4i*)g,
          (__attribute__((address_space(3))) v4i*)l,
          /*imm offset*/ 0, /*cpol*/ 0);
    }
  };

  if (c < n_chunks) issue(c, 0);

  for (; c < n_chunks; c += cstride) {
    const long long nxt = c + cstride;
    if (nxt < n_chunks) {
      issue(nxt, buf ^ 1);     // 4 more in flight on top of current 4
      WAIT_ASYNCCNT(UNR);      // in-order completion: current stage done
    } else {
      WAIT_ASYNCCNT(0);
    }
    asm volatile("" ::: "memory");  // keep LDS reads below the wait

    float* obase = out + c * (long long)CHUNK;
#pragma unroll
    for (int u = 0; u < UNR; ++u) {
      v4f v = lds[buf * (TPB * UNR) + u * TPB + tid];  // ds_load_b128
      v4f q = steq4(v, l0, l1, l2, l3);
      __builtin_nontemporal_store(q, (v4f*)(obase + (u * TPB + tid) * 4));
    }
    buf ^= 1;
  }
}
#endif  // HAVE_ASYNC_LDS

// Fallback / alternate path: grid-stride b128 non-temporal load/store with
// global_prefetch_b8 for the next stride. Always compiled.
__global__ __launch_bounds__(TPB) void steq_direct_kernel(
    const float* __restrict__ x, const float* __restrict__ levels,
    float* __restrict__ out, long long n4) {
  const float l0 = levels[0], l1 = levels[1], l2 = levels[2], l3 = levels[3];
  long long i = (long long)blockIdx.x * TPB + threadIdx.x;
  const long long stride = (long long)gridDim.x * TPB;
  for (; i < n4; i += stride) {
    __builtin_prefetch(x + (i + stride) * 4, 0, 0);  // speculative, safe if OOB
    v4f v = __builtin_nontemporal_load((const v4f*)x + i);
    v4f q = steq4(v, l0, l1, l2, l3);
    __builtin_nontemporal_store(q, (v4f*)out + i);
  }
}

// Scalar tail for elements not covered by the vector/chunk kernels.
__global__ void steq_tail_kernel(const float* __restrict__ x,
                                 const float* __restrict__ levels,
                                 float* __restrict__ out,
                                 long long start, long long n) {
  long long i = start + (long long)blockIdx.x * blockDim.x + threadIdx.x;
  if (i < n) out[i] = steq(x[i], levels[0], levels[1], levels[2], levels[3]);
}

extern "C" void kernel_launch(void* const* d_in, const int* in_sizes, int n_in,
                              void* d_out, int out_size, void* d_ws, size_t ws_size,
                              hipStream_t stream) {
  (void)n_in; (void)d_ws; (void)ws_size; (void)out_size;
  const float* x = (const float*)d_in[0];
  const float* levels = (const float*)d_in[1];
  float* out = (float*)d_out;
  const long long n = (long long)in_sizes[0];

  long long done = 0;
#if HAVE_ASYNC_LDS
  const long long n_chunks = n / CHUNK;
  if (n_chunks > 0) {
    int grid = (int)(n_chunks < 4096 ? n_chunks : 4096);  // >=4 stages/block at full size
    steq_async_kernel<<<grid, TPB, 0, stream>>>(x, levels, out, n_chunks);
  }
  done = n_chunks * (long long)CHUNK;
#else
  const long long n4 = n / 4;
  if (n4 > 0) {
    long long g = (n4 + TPB - 1) / TPB;
    int grid = (int)(g < 8192 ? g : 8192);
    steq_direct_kernel<<<grid, TPB, 0, stream>>>(x, levels, out, n4);
  }
  done = n4 * 4;
#endif

  const long long rem = n - done;
  if (rem > 0) {
    int grid = (int)((rem + 255) / 256);
    steq_tail_kernel<<<grid, 256, 0, stream>>>(x, levels, out, done, n);
  }
}